// PrunedHighOrderAggregator_34849364640474
// MI455X (gfx1250) — compile-verified
//
#include <hip/hip_runtime.h>

typedef __attribute__((ext_vector_type(2))) float v2f;
typedef __attribute__((ext_vector_type(8))) float v8f;
typedef __attribute__((ext_vector_type(4))) int   v4i;

#define N_NODES 50000
#define N_EDGES 800000
#define D_IN    256
#define D_OUT   128
#define EPS     1e-9f

#define AS_GLOBAL __attribute__((address_space(1)))
#define AS_LOCAL  __attribute__((address_space(3)))

// ---- native f32 global atomic add (no CAS fallback) -------------------------
__device__ __forceinline__ void atomic_add_f32(float* p, float v) {
#if __has_builtin(__builtin_amdgcn_global_atomic_fadd_f32)
    __builtin_amdgcn_global_atomic_fadd_f32((AS_GLOBAL float*)p, v);
#else
    unsafeAtomicAdd(p, v);   // HIP header: always the hardware global_atomic_add_f32
#endif
}

// ---- async global->LDS copy (CDNA5 ASYNCcnt path), with safe fallback -------
#if __has_builtin(__builtin_amdgcn_global_load_async_to_lds_b128)
#define HAVE_ASYNC_LDS 1
#else
#define HAVE_ASYNC_LDS 0
#endif

__device__ __forceinline__ void wait_asynccnt0() {
#if __has_builtin(__builtin_amdgcn_s_wait_asynccnt)
    __builtin_amdgcn_s_wait_asynccnt(0);
#else
    asm volatile("s_wait_asynccnt 0x0" ::: "memory");
#endif
}

// ---------------- zero the hop1 workspace (harness poisons d_ws) -------------
__global__ __launch_bounds__(256) void zero_ws_kernel(float4* __restrict__ p, int n4) {
    int i = blockIdx.x * 256 + threadIdx.x;
    if (i < n4) {
        float4 z = {0.f, 0.f, 0.f, 0.f};
        p[i] = z;
    }
}

// ---------------- SpMM: hop1[r] += val * feat[c], one wave32 per edge --------
// feat (51.2MB) and hop1 (51.2MB) both fit in the 192MB L2, so the gathers and
// the f32 atomic adds resolve on-chip. 256 floats/row / 32 lanes = 2x float4.
__global__ __launch_bounds__(256) void spmm_kernel(
    const float* __restrict__ feat,
    const int*   __restrict__ erow,
    const int*   __restrict__ ecol,
    const float* __restrict__ eval,
    float*       __restrict__ hop,
    int nE)
{
    int wave = (blockIdx.x * 256 + threadIdx.x) >> 5;
    int lane = threadIdx.x & 31;
    if (wave >= nE) return;

    int   r = erow[wave];
    int   c = ecol[wave];
    float v = eval[wave];

    const float4* src = (const float4*)(feat + (size_t)c * D_IN);
    float*        dst = hop + (size_t)r * D_IN;

#pragma unroll
    for (int i = 0; i < 2; ++i) {
        float4 x = src[lane + 32 * i];       // global_load_b128 (L2 hit)
        int    b = (lane + 32 * i) * 4;
        atomic_add_f32(dst + b + 0, v * x.x);
        atomic_add_f32(dst + b + 1, v * x.y);
        atomic_add_f32(dst + b + 2, v * x.z);
        atomic_add_f32(dst + b + 3, v * x.w);
    }
}

// ---------------- fused GEMM + bias + ReLU + LayerNorm + concat --------------
// grid = (3125, 2): blockIdx.x = 16-row tile, blockIdx.y = source (feat/hop1).
// 256 threads = 8 waves; wave w computes output columns [16w, 16w+16) via
// chained V_WMMA_F32_16X16X4_F32 (exact f32 math; GEMM FLOPs are negligible,
// so no precision drop is needed).
#define XPAD 4   // A-tile row stride 260 floats = 1040B: 16B-aligned for async
                 // b128 LDS writes, and 260 % 64 == 4 -> conflict-free column reads

__global__ __launch_bounds__(256) void feat_trans_fused_kernel(
    const float* __restrict__ feat, const float* __restrict__ hop,
    const float* __restrict__ W0, const float* __restrict__ b0,
    const float* __restrict__ sc0, const float* __restrict__ of0,
    const float* __restrict__ W1, const float* __restrict__ b1,
    const float* __restrict__ sc1, const float* __restrict__ of1,
    float* __restrict__ out)
{
    __shared__ float lx[16][D_IN + XPAD]; // staged A tile (~16.6KB)
    __shared__ float lh[16][D_OUT + 1];   // post-ReLU h tile
    __shared__ float ps[16][16];          // per-row partial sums
    __shared__ float ps2[16][16];         // per-row partial sums of squares
    __shared__ float mean_s[16];
    __shared__ float rstd_s[16];

    const int tile = blockIdx.x;
    const int s    = blockIdx.y;          // 0: feat->W0 (cols 0..127), 1: hop1->W1 (cols 128..255)
    const int row0 = tile * 16;           // N_NODES == 3125*16 exactly
    const int t    = threadIdx.x;
    const int wv   = t >> 5;
    const int lane = t & 31;

    const float* X   = s ? hop : feat;
    const float* W   = s ? W1  : W0;
    const float* bia = s ? b1  : b0;
    const float* sc  = s ? sc1 : sc0;
    const float* of  = s ? of1 : of0;

    // ---- stage 16x256 A tile into LDS ----
#if HAVE_ASYNC_LDS
    // 1024 16B chunks, 4 per thread: global_load_async_to_lds_b128 (ASYNCcnt)
#pragma unroll
    for (int i = 0; i < 4; ++i) {
        int cid = i * 256 + t;
        int r = cid >> 6, c4 = cid & 63;           // 64 chunks per 256-float row
        AS_GLOBAL v4i* g = (AS_GLOBAL v4i*)(X + (size_t)(row0 + r) * D_IN + c4 * 4);
        AS_LOCAL  v4i* l = (AS_LOCAL  v4i*)&lx[r][c4 * 4];
        __builtin_amdgcn_global_load_async_to_lds_b128(g, l, 0, 0);
    }
    wait_asynccnt0();
#else
#pragma unroll
    for (int i = 0; i < 16; ++i) {
        int idx = i * 256 + t;
        int r = idx >> 8, c = idx & 255;
        lx[r][c] = X[(size_t)(row0 + r) * D_IN + c];
    }
#endif
    __syncthreads();

    // ---- WMMA: 64 x V_WMMA_F32_16X16X4_F32, K = 256 ----
    // 32-bit A layout (ISA 7.12.2): lanes 0-15 hold {K,K+1}, lanes 16-31 hold {K+2,K+3}.
    const int n0   = wv * 16;
    const int half = lane >> 4;           // 0 or 1
    const int l16  = lane & 15;
    v8f acc = {};
#pragma unroll 4
    for (int k = 0; k < D_IN; k += 4) {
        v2f a, b;
        a.x = lx[l16][k + 2 * half];                            // ds_load_2addr_b32 pairs
        a.y = lx[l16][k + 2 * half + 1];
        b.x = W[(size_t)(k + 2 * half)     * D_OUT + n0 + l16]; // L2-resident W columns
        b.y = W[(size_t)(k + 2 * half + 1) * D_OUT + n0 + l16];
        acc = __builtin_amdgcn_wmma_f32_16x16x4_f32(
            /*neg_a=*/false, a, /*neg_b=*/false, b,
            /*c_mod=*/(short)0, acc, /*reuse_a=*/false, /*reuse_b=*/false);
    }

    // ---- bias + ReLU, scatter C (VGPR j, lane l -> row j + 8*(l>=16), col l&15) ----
    float bb = bia[n0 + l16];
#pragma unroll
    for (int j = 0; j < 8; ++j) {
        int   m = j + half * 8;
        float h = acc[j] + bb;
        h = h > 0.f ? h : 0.f;
        lh[m][n0 + l16] = h;
    }
    __syncthreads();

    // ---- row mean/var over 128 cols: 16 threads/row x 8 cols each ----
    {
        int r = t >> 4, seg = t & 15;
        float sum = 0.f, sq = 0.f;
#pragma unroll
        for (int c = 0; c < 8; ++c) {
            float h = lh[r][seg * 8 + c];
            sum += h; sq += h * h;
        }
        ps[r][seg] = sum; ps2[r][seg] = sq;
    }
    __syncthreads();
    if (t < 16) {
        float sum = 0.f, sq = 0.f;
#pragma unroll
        for (int i = 0; i < 16; ++i) { sum += ps[t][i]; sq += ps2[t][i]; }
        float mean = sum * (1.f / D_OUT);
        float var  = sq  * (1.f / D_OUT) - mean * mean + EPS;  // population variance
        mean_s[t] = mean;
        rstd_s[t] = rsqrtf(var);
    }
    __syncthreads();

    // ---- normalize + concat-write: out[row][s*128 + c], 8 elems/thread ----
#pragma unroll
    for (int i = 0; i < 8; ++i) {
        int idx = i * 256 + t;
        int r = idx >> 7, c = idx & 127;
        float h = lh[r][c];
        float y = (h - mean_s[r]) * sc[c] * rstd_s[r] + of[c];
        out[(size_t)(row0 + r) * (2 * D_OUT) + s * D_OUT + c] = y;
    }
}

// ----------------------------- launcher --------------------------------------
extern "C" void kernel_launch(void* const* d_in, const int* in_sizes, int n_in,
                              void* d_out, int out_size, void* d_ws, size_t ws_size,
                              hipStream_t stream) {
    const float* feat = (const float*)d_in[0];
    const float* W0   = (const float*)d_in[1];
    const float* b0   = (const float*)d_in[2];
    const float* sc0  = (const float*)d_in[3];
    const float* of0  = (const float*)d_in[4];
    const float* W1   = (const float*)d_in[5];
    const float* b1   = (const float*)d_in[6];
    const float* sc1  = (const float*)d_in[7];
    const float* of1  = (const float*)d_in[8];
    const int*   erow = (const int*)d_in[9];
    const int*   ecol = (const int*)d_in[10];
    const float* eval = (const float*)d_in[11];
    float*       out  = (float*)d_out;
    float*       hop  = (float*)d_ws;   // 50000*256*4 = 51.2 MB scratch

    // 1) zero hop1
    int n4 = (N_NODES * D_IN) / 4;
    zero_ws_kernel<<<(n4 + 255) / 256, 256, 0, stream>>>((float4*)hop, n4);

    // 2) SpMM via L2 f32 atomics: one wave per edge, 8 waves per block
    spmm_kernel<<<(N_EDGES + 7) / 8, 256, 0, stream>>>(feat, erow, ecol, eval, hop, N_EDGES);

    // 3) fused dual GEMM + ReLU + LayerNorm + concat
    dim3 grid(N_NODES / 16, 2);
    feat_trans_fused_kernel<<<grid, 256, 0, stream>>>(
        feat, hop, W0, b0, sc0, of0, W1, b1, sc1, of1, out);
}